// LinearAttention_4148938408225
// MI455X (gfx1250) — compile-verified
//
#include <hip/hip_runtime.h>
#include <hip/hip_bf16.h>
#include <math.h>

// ---------------- problem constants ----------------
#define DIMC   128        // channels
#define NSEQ   16384      // sequence length
#define BATCH  8
#define HEADS  4
#define DH     32
#define TN     128        // columns (n) per tile
#define NPER   2          // n-tiles per block (double-buffered)
#define NCT    8          // 8 col-subtiles of 16
#define PW     132        // LDS pitch for 128-wide rows (bank-conflict pad, mult of 4)
#define PQ     20         // LDS pitch for 16-wide staging (mult of 4)

typedef __attribute__((ext_vector_type(2))) float v2f;
typedef __attribute__((ext_vector_type(8))) float v8f;
typedef __attribute__((ext_vector_type(4))) int   v4i;

// pointer-to-AS1(v4i) and pointer-to-AS3(v4i) for the async builtin
typedef __attribute__((address_space(1))) v4i* gptr_v4i;
typedef __attribute__((address_space(3))) v4i* lptr_v4i;

// D = A(16x4 f32) * B(4x16 f32) + C(16x16 f32), wave32 CDNA5 WMMA.
__device__ __forceinline__ v8f wmma4(v2f a, v2f b, v8f c) {
    return __builtin_amdgcn_wmma_f32_16x16x4_f32(
        /*neg_a=*/false, a, /*neg_b=*/false, b,
        /*c_mod=*/(short)0, c, /*reuse_a=*/false, /*reuse_b=*/false);
}

// ---- CDNA5 async global->LDS (ASYNCcnt) with safe fallback ----
#ifdef __has_builtin
#  if __has_builtin(__builtin_amdgcn_global_load_async_to_lds_b128)
#    define HAVE_ASYNC_B128 1
#  endif
#  if __has_builtin(__builtin_amdgcn_s_wait_asynccnt)
#    define HAVE_WAIT_ASYNC 1
#  endif
#endif

__device__ __forceinline__ void g2l_b128(float* lds_dst, const float* gsrc) {
#if defined(HAVE_ASYNC_B128)
    __builtin_amdgcn_global_load_async_to_lds_b128(
        (gptr_v4i)(gsrc),
        (lptr_v4i)(lds_dst),
        /*imm offset=*/0, /*cpol=*/0);
#else
    *(float4*)lds_dst = *(const float4*)gsrc;
#endif
}

__device__ __forceinline__ void wait_async_lds() {
#if defined(HAVE_ASYNC_B128)
#  if defined(HAVE_WAIT_ASYNC)
    __builtin_amdgcn_s_wait_asynccnt(0);
#  else
    asm volatile("s_wait_asynccnt 0" ::: "memory");
#  endif
#endif
}

// ---------------- pass 0: zero accumulators ----------------
__global__ void k_zero(float* __restrict__ p, int n) {
    int i = blockIdx.x * 256 + threadIdx.x;
    if (i < n) p[i] = 0.0f;
}

// ---------------- pass 1: k/v GEMM + exp(k) + context/Z partials ----------------
// grid (NSEQ/(TN*NPER), BATCH), 256 threads, dynamic LDS
__global__ __launch_bounds__(256) void k_pass1(const float* __restrict__ x,
                                               const float* __restrict__ w_qkv,
                                               float* __restrict__ Z,
                                               float* __restrict__ ctx) {
    extern __shared__ float smem[];
    float* sW  = smem;                  // [256][PW]  w_qkv rows 128..383 (k then v)
    float* sX0 = sW + 256 * PW;         // [128][PW]  x tile (buffer 0)
    float* sX1 = sX0 + 128 * PW;        // [128][PW]  x tile (buffer 1)
    float* sQ  = sX1 + 128 * PW;        // [256][PQ]  staging: rows 0..127=k, 128..255=v

    const int t    = threadIdx.x;
    const int b    = blockIdx.y;
    const int n0b  = blockIdx.x * (TN * NPER);
    const int lane = t & 31;
    const int wv   = t >> 5;            // wave id 0..7
    const int lo   = lane & 15;
    const int hi   = lane >> 4;
    const float* xb = x + (size_t)b * DIMC * NSEQ;

    // ---- async-load w_qkv rows 128..383 (8192 float4) + x tile 0 ----
    #pragma unroll
    for (int i = 0; i < 32; ++i) {
        int e4  = t + i * 256;
        int row = e4 >> 5;              // 32 float4 per row
        int c4  = e4 & 31;
        g2l_b128(sW + row * PW + c4 * 4, w_qkv + 128 * DIMC + e4 * 4);
    }
    #pragma unroll
    for (int i = 0; i < 16; ++i) {
        int e4  = t + i * 256;
        int row = e4 >> 5;
        int c4  = e4 & 31;
        g2l_b128(sX0 + row * PW + c4 * 4, xb + (size_t)row * NSEQ + n0b + c4 * 4);
    }
    wait_async_lds();
    __syncthreads();

    float zacc = 0.0f;
    float cacc[16];
    #pragma unroll
    for (int i = 0; i < 16; ++i) cacc[i] = 0.0f;
    const int p_    = t & 127;          // (h,d) owner for ctx phase
    const int hh    = p_ >> 5;
    const int ehalf = t >> 7;           // e range: [ehalf*16, ehalf*16+16)

    for (int tile = 0; tile < NPER; ++tile) {
        float* sXc = (tile & 1) ? sX1 : sX0;
        float* sXn = (tile & 1) ? sX0 : sX1;
        // ---- prefetch next tile asynchronously while computing this one ----
        if (tile + 1 < NPER) {
            const int n1 = n0b + (tile + 1) * TN;
            #pragma unroll
            for (int i = 0; i < 16; ++i) {
                int e4  = t + i * 256;
                int row = e4 >> 5;
                int c4  = e4 & 31;
                g2l_b128(sXn + row * PW + c4 * 4, xb + (size_t)row * NSEQ + n1 + c4 * 4);
            }
        }

        for (int ct = 0; ct < NCT; ++ct) {
            const int cb = ct * 16;
            // ---- GEMM: wave handles kv row-tiles wv and wv+8 ----
            v8f acc0 = {};
            v8f acc1 = {};
            const float* A0 = sW + (wv * 16 + lo) * PW;
            const float* A1 = sW + ((wv + 8) * 16 + lo) * PW;
            #pragma unroll 4
            for (int k = 0; k < DIMC; k += 4) {
                const int ka = k + 2 * hi;
                v2f a0 = *(const v2f*)(A0 + ka);
                v2f a1 = *(const v2f*)(A1 + ka);
                v2f bb;
                bb.x = sXc[ka * PW + cb + lo];
                bb.y = sXc[(ka + 1) * PW + cb + lo];
                acc0 = wmma4(a0, bb, acc0);
                acc1 = wmma4(a1, bb, acc1);
            }
            // ---- epilogue: C/D layout -> LDS staging ----
            #pragma unroll
            for (int r = 0; r < 8; ++r) {
                sQ[(wv * 16 + r + 8 * hi) * PQ + lo]       = acc0[r];
                sQ[((wv + 8) * 16 + r + 8 * hi) * PQ + lo] = acc1[r];
            }
            __syncthreads();
            // ---- phase A: exp(k), Z partial (rows 0..127) ----
            if (t < 128) {
                float* row = sQ + t * PQ;
                #pragma unroll
                for (int c = 0; c < 16; ++c) {
                    float e = __expf(row[c]);
                    row[c] = e;
                    zacc += e;
                }
            }
            __syncthreads();
            // ---- phase B: ctx[h][d][e] += sum_cols expk[d]*v[e] ----
            {
                const float* krow = sQ + p_ * PQ;
                float4 k0 = *(const float4*)(krow + 0);
                float4 k1 = *(const float4*)(krow + 4);
                float4 k2 = *(const float4*)(krow + 8);
                float4 k3 = *(const float4*)(krow + 12);
                #pragma unroll
                for (int e = 0; e < 16; ++e) {
                    const float* vr = sQ + (128 + hh * DH + ehalf * 16 + e) * PQ;
                    float4 v0 = *(const float4*)(vr + 0);
                    float4 v1 = *(const float4*)(vr + 4);
                    float4 v2 = *(const float4*)(vr + 8);
                    float4 v3 = *(const float4*)(vr + 12);
                    float s = k0.x * v0.x + k0.y * v0.y + k0.z * v0.z + k0.w * v0.w
                            + k1.x * v1.x + k1.y * v1.y + k1.z * v1.z + k1.w * v1.w
                            + k2.x * v2.x + k2.y * v2.y + k2.z * v2.z + k2.w * v2.w
                            + k3.x * v3.x + k3.y * v3.y + k3.z * v3.z + k3.w * v3.w;
                    cacc[e] += s;
                }
            }
            __syncthreads();
        }
        wait_async_lds();
        __syncthreads();
    }
    // ---- flush partials ----
    if (t < 128) atomicAdd(&Z[b * DIMC + t], zacc);
    {
        const int dd = p_ & 31;
        float* cbase = ctx + (((size_t)b * HEADS + hh) * DH + dd) * DH + ehalf * 16;
        #pragma unroll
        for (int e = 0; e < 16; ++e) atomicAdd(&cbase[e], cacc[e]);
    }
}

// ---------------- pass 2: W2[b] = w_out * blockdiag(ctx/Z) ----------------
// grid (BATCH), 256 threads
__global__ __launch_bounds__(256) void k_pass2(const float* __restrict__ w_out,
                                               const float* __restrict__ ctx,
                                               const float* __restrict__ Z,
                                               float* __restrict__ W2) {
    const int b = blockIdx.x;
    const int t = threadIdx.x;
    #pragma unroll 4
    for (int i = 0; i < 64; ++i) {
        int idx  = t + i * 256;               // 128*128 entries
        int o    = idx >> 7;
        int cd   = idx & 127;                 // = h*32 + d
        int h    = cd >> 5;
        const float* wr = w_out + o * DIMC + h * DH;
        const float* cr = ctx + (((size_t)b * HEADS + h) * DH + (cd & 31)) * DH;
        float s = 0.0f;
        #pragma unroll
        for (int e = 0; e < DH; ++e) s += wr[e] * cr[e];
        float z = Z[b * DIMC + cd];
        W2[(size_t)b * DIMC * DIMC + idx] = s / fmaxf(z, 1e-30f);
    }
}

// ---------------- pass 3: q GEMM -> softmax(d) -> W2 GEMM -> bias -> RMSNorm ----------------
// grid (NSEQ/(TN*NPER), BATCH), 256 threads, dynamic LDS
__global__ __launch_bounds__(256) void k_pass3(const float* __restrict__ x,
                                               const float* __restrict__ w_qkv,
                                               const float* __restrict__ W2,
                                               const float* __restrict__ b_out,
                                               const float* __restrict__ g,
                                               float* __restrict__ out) {
    extern __shared__ float smem[];
    float* sWq  = smem;                 // [128][PW] w_qkv rows 0..127 (q proj)
    float* sW2  = sWq + 128 * PW;       // [128][PW]
    float* sX0  = sW2 + 128 * PW;       // [128][PW]
    float* sX1  = sX0 + 128 * PW;       // [128][PW]
    float* sQ   = sX1 + 128 * PW;       // [128][PQ]
    float* sO   = sQ + 128 * PQ;        // [128][PQ]
    float* sP   = sO + 128 * PQ;        // [16 segs][16 cols]
    float* sNrm = sP + 256;             // [16]

    const int t    = threadIdx.x;
    const int b    = blockIdx.y;
    const int n0b  = blockIdx.x * (TN * NPER);
    const int lane = t & 31;
    const int wv   = t >> 5;
    const int lo   = lane & 15;
    const int hi   = lane >> 4;
    const float scale = 0.17677669529663687f;   // 32^-0.5
    const float s128  = 11.313708498984761f;    // sqrt(128)
    const float* xb = x + (size_t)b * DIMC * NSEQ;

    {   // async-load q-projection weights, W2[b], x tile 0
        const float* w2 = W2 + (size_t)b * DIMC * DIMC;
        #pragma unroll
        for (int i = 0; i < 16; ++i) {
            int e4  = t + i * 256;
            int row = e4 >> 5;
            int c4  = e4 & 31;
            g2l_b128(sWq + row * PW + c4 * 4, w_qkv + e4 * 4);
            g2l_b128(sW2 + row * PW + c4 * 4, w2 + e4 * 4);
            g2l_b128(sX0 + row * PW + c4 * 4, xb + (size_t)row * NSEQ + n0b + c4 * 4);
        }
    }
    wait_async_lds();
    __syncthreads();

    for (int tile = 0; tile < NPER; ++tile) {
        float* sXc = (tile & 1) ? sX1 : sX0;
        float* sXn = (tile & 1) ? sX0 : sX1;
        const int n0 = n0b + tile * TN;
        if (tile + 1 < NPER) {
            const int n1 = n0b + (tile + 1) * TN;
            #pragma unroll
            for (int i = 0; i < 16; ++i) {
                int e4  = t + i * 256;
                int row = e4 >> 5;
                int c4  = e4 & 31;
                g2l_b128(sXn + row * PW + c4 * 4, xb + (size_t)row * NSEQ + n1 + c4 * 4);
            }
        }

        for (int ct = 0; ct < NCT; ++ct) {
            const int cb = ct * 16;
            // ---- GEMM1: q = Wq @ x_sub  (wave wv owns row-tile wv) ----
            {
                v8f acc = {};
                const float* A = sWq + (wv * 16 + lo) * PW;
                #pragma unroll 4
                for (int k = 0; k < DIMC; k += 4) {
                    const int ka = k + 2 * hi;
                    v2f a = *(const v2f*)(A + ka);
                    v2f bb;
                    bb.x = sXc[ka * PW + cb + lo];
                    bb.y = sXc[(ka + 1) * PW + cb + lo];
                    acc = wmma4(a, bb, acc);
                }
                #pragma unroll
                for (int r = 0; r < 8; ++r)
                    sQ[(wv * 16 + r + 8 * hi) * PQ + lo] = acc[r];
            }
            __syncthreads();
            // ---- softmax over d (32 rows per head) per column, * scale ----
            if (t < 64) {
                int h = t >> 4, c = t & 15;
                float* colp = sQ + (h * DH) * PQ + c;
                float m = -1e30f;
                #pragma unroll
                for (int d = 0; d < DH; ++d) m = fmaxf(m, colp[d * PQ]);
                float s = 0.0f;
                #pragma unroll
                for (int d = 0; d < DH; ++d) {
                    float e = __expf(colp[d * PQ] - m);
                    colp[d * PQ] = e;
                    s += e;
                }
                float inv = scale / s;
                #pragma unroll
                for (int d = 0; d < DH; ++d) colp[d * PQ] *= inv;
            }
            __syncthreads();
            // ---- GEMM2: y = W2 @ q_sub, + bias ----
            {
                v8f acc = {};
                const float* A = sW2 + (wv * 16 + lo) * PW;
                #pragma unroll 4
                for (int k = 0; k < DIMC; k += 4) {
                    const int ka = k + 2 * hi;
                    v2f a = *(const v2f*)(A + ka);
                    v2f bb;
                    bb.x = sQ[ka * PQ + lo];
                    bb.y = sQ[(ka + 1) * PQ + lo];
                    acc = wmma4(a, bb, acc);
                }
                #pragma unroll
                for (int r = 0; r < 8; ++r) {
                    int row = wv * 16 + r + 8 * hi;
                    sO[row * PQ + lo] = acc[r] + b_out[row];
                }
            }
            __syncthreads();
            // ---- RMSNorm over channels: partial sums of squares ----
            {
                int c = t & 15, seg = t >> 4;
                float ss = 0.0f;
                #pragma unroll
                for (int r = 0; r < 8; ++r) {
                    float v = sO[(seg * 8 + r) * PQ + c];
                    ss += v * v;
                }
                sP[seg * 16 + c] = ss;
            }
            __syncthreads();
            if (t < 16) {
                float ss = 0.0f;
                #pragma unroll
                for (int sgi = 0; sgi < 16; ++sgi) ss += sP[sgi * 16 + t];
                sNrm[t] = s128 / fmaxf(sqrtf(ss), 1e-12f);
            }
            __syncthreads();
            // ---- scaled writeout ----
            {
                float* ob = out + (size_t)b * DIMC * NSEQ + n0 + cb;
                #pragma unroll
                for (int i = 0; i < 8; ++i) {
                    int e   = t + i * 256;
                    int row = e >> 4;
                    int c   = e & 15;
                    ob[(size_t)row * NSEQ + c] = sO[row * PQ + c] * sNrm[c] * g[row];
                }
            }
            __syncthreads();
        }
        wait_async_lds();
        __syncthreads();
    }
}

// ---------------- launcher ----------------
extern "C" void kernel_launch(void* const* d_in, const int* in_sizes, int n_in,
                              void* d_out, int out_size, void* d_ws, size_t ws_size,
                              hipStream_t stream) {
    const float* x     = (const float*)d_in[0];
    const float* w_qkv = (const float*)d_in[1];
    const float* w_out = (const float*)d_in[2];
    const float* b_out = (const float*)d_in[3];
    const float* g     = (const float*)d_in[4];
    float* out = (float*)d_out;

    float* Z   = (float*)d_ws;                      // [B][128]
    float* ctx = Z + BATCH * DIMC;                  // [B][H][DH][DH]
    float* W2  = ctx + BATCH * HEADS * DH * DH;     // [B][128][128]

    const int nzero = BATCH * DIMC + BATCH * HEADS * DH * DH;   // Z + ctx
    k_zero<<<dim3((nzero + 255) / 256), 256, 0, stream>>>(Z, nzero);

    const size_t lds1 = (size_t)(256 * PW + 2 * 128 * PW + 256 * PQ) * sizeof(float);
    k_pass1<<<dim3(NSEQ / (TN * NPER), BATCH), 256, lds1, stream>>>(x, w_qkv, Z, ctx);

    k_pass2<<<dim3(BATCH), 256, 0, stream>>>(w_out, ctx, Z, W2);

    const size_t lds3 = (size_t)(4 * 128 * PW + 2 * 128 * PQ + 256 + 16) * sizeof(float);
    k_pass3<<<dim3(NSEQ / (TN * NPER), BATCH), 256, lds3, stream>>>(x, w_qkv, W2, b_out, g, out);
}